// RVQRefiner_36575941493555
// MI455X (gfx1250) — compile-verified
//
#include <hip/hip_runtime.h>
#include <hip/hip_bf16.h>

typedef float v2f __attribute__((ext_vector_type(2)));
typedef float v8f __attribute__((ext_vector_type(8)));

#define D_MODEL 512
#define K_CODES 1024
#define N_LEVELS 8
#define TOK_WG 64          // tokens per workgroup
#define RS (D_MODEL + 4)   // padded LDS row stride (rows land on distinct banks)
#define THREADS 256

// ---------------------------------------------------------------------------
// Kernel 1: c2ws[l*K + k] = 0.5 * sum_d cb[l][k][d]^2   (one row per wave)
// ---------------------------------------------------------------------------
__global__ __launch_bounds__(THREADS) void rvq_c2_kernel(
    const float* __restrict__ cb, float* __restrict__ c2ws) {
  const int lane = threadIdx.x & 31;
  const int wave = (blockIdx.x * THREADS + threadIdx.x) >> 5;  // 0..8191
  const float* row = cb + (size_t)wave * D_MODEL;
  float s = 0.0f;
  #pragma unroll 4
  for (int i = lane; i < D_MODEL; i += 32) {
    float v = row[i];
    s += v * v;
  }
  #pragma unroll
  for (int off = 16; off > 0; off >>= 1) s += __shfl_xor(s, off, 32);
  if (lane == 0) c2ws[wave] = 0.5f * s;
}

// ---------------------------------------------------------------------------
// Kernel 2: fused residual VQ over all 8 levels.
//   Block: 256 threads = 8 waves. Waves 2g,2g+1 share token group g (16 tok),
//   splitting the 1024 codes in half. Residual tile lives in LDS end-to-end.
//   WMMA mapping: A = codebook (M=16 codes x K=4 dims), B = residual tokens
//   (K=4 dims x N=16 tokens), C/D = scores (codes x tokens), f32 throughout.
// ---------------------------------------------------------------------------
__global__ __launch_bounds__(THREADS) void rvq_main_kernel(
    const float* __restrict__ x, const float* __restrict__ cb,
    const float* __restrict__ c2ws, float* __restrict__ outQ,
    int* __restrict__ outI) {
  extern __shared__ float smem[];
  float* sResid = smem;                     // TOK_WG * RS
  float* sC2    = sResid + TOK_WG * RS;     // K_CODES
  float* sBS    = sC2 + K_CODES;            // 8 waves * 16 tokens
  int*   sBI    = (int*)(sBS + 8 * 16);     // 8 * 16
  int*   sIdx   = sBI + 8 * 16;             // TOK_WG

  const int tid  = threadIdx.x;
  const int lane = tid & 31;
  const int w    = tid >> 5;       // wave 0..7
  const int g    = w >> 1;         // token group 0..3
  const int half = w & 1;          // code half: [0,512) or [512,1024)
  const size_t tokBase = (size_t)blockIdx.x * TOK_WG;

  const int lr   = lane & 15;            // row within 16 (code row / token col)
  const int koff = (lane >> 4) << 1;     // K-offset: lanes 0-15 -> 0, 16-31 -> 2
  const int lco  = (lane >> 4) << 3;     // +8 code offset for upper half-wave

  // Stage residual tile = x tile (coalesced).
  for (int i = tid; i < TOK_WG * D_MODEL; i += THREADS) {
    int t = i >> 9, d = i & (D_MODEL - 1);
    sResid[t * RS + d] = x[(tokBase + t) * D_MODEL + d];
  }

  const float* bRow = sResid + (g * 16 + lr) * RS + koff;

  for (int lvl = 0; lvl < N_LEVELS; ++lvl) {
    // Stage this level's 0.5*|c|^2 (L2-resident, 4KB).
    for (int i = tid; i < K_CODES; i += THREADS) sC2[i] = c2ws[lvl * K_CODES + i];
    __syncthreads();

    float best = -3.402823466e38f;
    int bestIdx = 0;

    // Each wave scans 512 codes as 8 passes of 4 code-tiles (64 codes/pass).
    for (int outer = 0; outer < 8; ++outer) {
      const int codeBase = half * 512 + outer * 64;
      const float* aPtr =
          cb + ((size_t)(lvl * K_CODES + codeBase + lr)) * D_MODEL + koff;
      // Light prefetch of the next pass's rows (global_prefetch_b8).
      if (outer < 7) {
        __builtin_prefetch(aPtr + (size_t)64 * D_MODEL, 0, 1);
        __builtin_prefetch(aPtr + (size_t)(64 + 16) * D_MODEL, 0, 1);
        __builtin_prefetch(aPtr + (size_t)(64 + 32) * D_MODEL, 0, 1);
        __builtin_prefetch(aPtr + (size_t)(64 + 48) * D_MODEL, 0, 1);
      }

      v8f c0 = {}, c1 = {}, c2v = {}, c3 = {};
      #pragma unroll 4
      for (int dB = 0; dB < D_MODEL; dB += 4) {
        v2f b  = *(const v2f*)(bRow + dB);                      // ds_load_b64
        v2f a0 = *(const v2f*)(aPtr + dB);                      // global b64 (L2 hot)
        v2f a1 = *(const v2f*)(aPtr + 16 * D_MODEL + dB);
        v2f a2 = *(const v2f*)(aPtr + 32 * D_MODEL + dB);
        v2f a3 = *(const v2f*)(aPtr + 48 * D_MODEL + dB);
        c0  = __builtin_amdgcn_wmma_f32_16x16x4_f32(false, a0, false, b, (short)0, c0,  false, false);
        c1  = __builtin_amdgcn_wmma_f32_16x16x4_f32(false, a1, false, b, (short)0, c1,  false, false);
        c2v = __builtin_amdgcn_wmma_f32_16x16x4_f32(false, a2, false, b, (short)0, c2v, false, false);
        c3  = __builtin_amdgcn_wmma_f32_16x16x4_f32(false, a3, false, b, (short)0, c3,  false, false);
      }

      // score = cross - 0.5*|c|^2 ; v-major scan over the 4 tiles + strict '>' keeps
      // the first-visited maximum, which within a pass need not be the lowest index.
      #pragma unroll
      for (int v = 0; v < 8; ++v) {
        int cix0 = codeBase + v + lco;
        float s0 = c0[v] - sC2[cix0];
        if (s0 > best) { best = s0; bestIdx = cix0; }
        int cix1 = cix0 + 16;
        float s1 = c1[v] - sC2[cix1];
        if (s1 > best) { best = s1; bestIdx = cix1; }
        int cix2 = cix0 + 32;
        float s2 = c2v[v] - sC2[cix2];
        if (s2 > best) { best = s2; bestIdx = cix2; }
        int cix3 = cix0 + 48;
        float s3 = c3[v] - sC2[cix3];
        if (s3 > best) { best = s3; bestIdx = cix3; }
      }
    }

    // Merge lane L with lane L+16 (they hold interleaved +0..7 / +8..15 codes
    // of the same token column); lower index wins ties.
    {
      float ob = __shfl_xor(best, 16, 32);
      int   oi = __shfl_xor(bestIdx, 16, 32);
      if (ob > best || (ob == best && oi < bestIdx)) { best = ob; bestIdx = oi; }
    }
    if (lane < 16) {
      sBS[w * 16 + lane] = best;
      sBI[w * 16 + lane] = bestIdx;
    }
    __syncthreads();

    // Merge the two waves of each token group; emit indices output.
    if (tid < TOK_WG) {
      int gg = tid >> 4, j = tid & 15;
      float s0 = sBS[(gg * 2) * 16 + j];
      int   i0 = sBI[(gg * 2) * 16 + j];
      float s1 = sBS[(gg * 2 + 1) * 16 + j];
      int   i1 = sBI[(gg * 2 + 1) * 16 + j];
      int win = (s1 > s0 || (s1 == s0 && i1 < i0)) ? i1 : i0;
      sIdx[tid] = win;
      outI[(tokBase + tid) * N_LEVELS + lvl] = win;
    }
    __syncthreads();

    // residual -= cb[idx]  (coalesced row gather; rows are L2-resident)
    for (int i = tid; i < TOK_WG * D_MODEL; i += THREADS) {
      int t = i >> 9, d = i & (D_MODEL - 1);
      sResid[t * RS + d] -=
          cb[((size_t)lvl * K_CODES + sIdx[t]) * D_MODEL + d];
    }
    __syncthreads();
  }

  // quantized = q_sum = x - residual_final  (straight-through forward value)
  for (int i = tid; i < TOK_WG * D_MODEL; i += THREADS) {
    int t = i >> 9, d = i & (D_MODEL - 1);
    size_t gi = (tokBase + t) * D_MODEL + d;
    outQ[gi] = x[gi] - sResid[t * RS + d];
  }
}

// ---------------------------------------------------------------------------
extern "C" void kernel_launch(void* const* d_in, const int* in_sizes, int n_in,
                              void* d_out, int out_size, void* d_ws,
                              size_t ws_size, hipStream_t stream) {
  const float* x  = (const float*)d_in[0];  // [8,4096,512]
  const float* cb = (const float*)d_in[1];  // [8,1024,512]
  float* outQ = (float*)d_out;
  int*   outI = (int*)((float*)d_out + (size_t)8 * 4096 * D_MODEL);
  float* c2ws = (float*)d_ws;               // 8*1024 floats

  const size_t smemBytes =
      (size_t)(TOK_WG * RS + K_CODES + 8 * 16) * sizeof(float) +
      (size_t)(8 * 16 + TOK_WG) * sizeof(int);
  (void)hipFuncSetAttribute((const void*)rvq_main_kernel,
                            hipFuncAttributeMaxDynamicSharedMemorySize,
                            (int)smemBytes);

  // 8192 codebook rows, one per wave, 8 waves/block -> 1024 blocks.
  rvq_c2_kernel<<<1024, THREADS, 0, stream>>>(cb, c2ws);

  // 32768 tokens / 64 per block -> 512 blocks.
  const int nTok = 8 * 4096;
  rvq_main_kernel<<<nTok / TOK_WG, THREADS, smemBytes, stream>>>(
      x, cb, c2ws, outQ, outI);
}